// MolecularGCN_103079215284
// MI455X (gfx1250) — compile-verified
//
#include <hip/hip_runtime.h>

// MolecularGCN fused kernel for gfx1250 (MI455X, wave32, WMMA).
// One workgroup (8 waves) per batch; adj_n recomputed from adjs (L2-resident),
// all intermediates (XW, H) live in LDS; fp32 WMMA 16x16x4 for all 4 GEMMs.

typedef float v2f __attribute__((ext_vector_type(2)));
typedef float v8f __attribute__((ext_vector_type(8)));

constexpr int NB  = 512;   // batches
constexpr int NN  = 256;   // nodes per graph
constexpr int AD  = 75;    // atom feature dim
constexpr int HD  = 128;   // hidden dim
constexpr int KP1 = 76;    // atoms K padded to mult-of-4; stride 76 is LDS-bank safe
constexpr int HS  = 132;   // LDS row stride for H (128+4): bank-safe column reads

// LDS plan (floats): bufA[256*132] | bufB[256*128] | sdis[256]  == 261 KB dynamic LDS
constexpr int OFF_B   = NN * HS;          // 33792
constexpr int OFF_DIS = OFF_B + NN * HD;  // 66560
constexpr int SMEM_FLOATS = OFF_DIS + NN; // 66816

__device__ __forceinline__ v8f wmma_f32(v2f a, v2f b, v8f c) {
    // D = A(16x4 f32) * B(4x16 f32) + C(16x16 f32)
    return __builtin_amdgcn_wmma_f32_16x16x4_f32(
        /*neg_a=*/false, a, /*neg_b=*/false, b,
        /*c_mod=*/(short)0, c, /*reuse_a=*/false, /*reuse_b=*/false);
}

// XW = X(LDS, 256 x kin, row stride xs) @ W(global, kin x 128) -> dstXW(LDS, stride 128)
__device__ __forceinline__ void gemm_xw(const float* srcX, int xs, int kin, int kloop,
                                        const float* __restrict__ W,
                                        float* dstXW, int wave, int lane)
{
    const int half = lane >> 4;   // 0: K pair {k,k+1}, 1: K pair {k+2,k+3}
    const int l15  = lane & 15;   // M (for A) / N (for B)
    const int M0   = wave * 32;   // this wave owns output rows [M0, M0+32)
    const v8f vzero = {0.f,0.f,0.f,0.f,0.f,0.f,0.f,0.f};
    v8f acc[2][8];
#pragma unroll
    for (int mt = 0; mt < 2; ++mt)
#pragma unroll
        for (int nt = 0; nt < 8; ++nt)
            acc[mt][nt] = vzero;

    for (int k0 = 0; k0 < kloop; k0 += 4) {
        const int ka = k0 + half * 2;
        v2f afrag[2];
#pragma unroll
        for (int mt = 0; mt < 2; ++mt) {
            const float* p = srcX + (M0 + mt * 16 + l15) * xs + ka;
            afrag[mt].x = p[0];
            afrag[mt].y = p[1];
        }
        v2f bfrag[8];
#pragma unroll
        for (int nt = 0; nt < 8; ++nt) {
            const int col = nt * 16 + l15;
            bfrag[nt].x = (ka     < kin) ? W[ka * HD + col]       : 0.f;
            bfrag[nt].y = (ka + 1 < kin) ? W[(ka + 1) * HD + col] : 0.f;
        }
#pragma unroll
        for (int mt = 0; mt < 2; ++mt)
#pragma unroll
            for (int nt = 0; nt < 8; ++nt)
                acc[mt][nt] = wmma_f32(afrag[mt], bfrag[nt], acc[mt][nt]);
    }
    // D layout: lanes 0-15: (M=v, N=l15); lanes 16-31: (M=v+8, N=l15)
#pragma unroll
    for (int mt = 0; mt < 2; ++mt)
#pragma unroll
        for (int nt = 0; nt < 8; ++nt) {
            const int col = nt * 16 + l15;
#pragma unroll
            for (int v = 0; v < 8; ++v) {
                const int row = M0 + mt * 16 + half * 8 + v;
                dstXW[row * HD + col] = acc[mt][nt][v];
            }
        }
}

// out = adj_n^T @ XW + bias, adj_n recomputed on the fly from global adjs + LDS dis.
// Writes to LDS (stride ldsStride, with relu) or to global.
__device__ __forceinline__ void gemm_adj(const float* __restrict__ adjB,
                                         const float* sdis, const float* srcXW,
                                         const float* __restrict__ bias,
                                         bool relu,
                                         float* dstLDS, int ldsStride,
                                         float* __restrict__ dstG, bool toGlobal,
                                         int wave, int lane)
{
    const int half = lane >> 4;
    const int l15  = lane & 15;
    const int M0   = wave * 32;
    const v8f vzero = {0.f,0.f,0.f,0.f,0.f,0.f,0.f,0.f};
    v8f acc[2][8];
#pragma unroll
    for (int mt = 0; mt < 2; ++mt)
#pragma unroll
        for (int nt = 0; nt < 8; ++nt)
            acc[mt][nt] = vzero;

    int   cc[2];
    float dc[2];
#pragma unroll
    for (int mt = 0; mt < 2; ++mt) {
        cc[mt] = M0 + mt * 16 + l15;   // output node index c
        dc[mt] = sdis[cc[mt]];
    }

    for (int k0 = 0; k0 < NN; k0 += 4) {
        const int   r0  = k0 + half * 2;
        const float dr0 = sdis[r0];
        const float dr1 = sdis[r0 + 1];
        v2f afrag[2];
#pragma unroll
        for (int mt = 0; mt < 2; ++mt) {
            // A^T[c, r] = adj_n[r, c] = ((adjs[r,c]>0.5)||(r==c)) * dis[r]*dis[c]
            const float v0 = adjB[r0 * NN + cc[mt]];
            const float v1 = adjB[(r0 + 1) * NN + cc[mt]];
            afrag[mt].x = ((v0 > 0.5f) || (r0     == cc[mt])) ? dr0 * dc[mt] : 0.f;
            afrag[mt].y = ((v1 > 0.5f) || (r0 + 1 == cc[mt])) ? dr1 * dc[mt] : 0.f;
        }
        v2f bfrag[8];
#pragma unroll
        for (int nt = 0; nt < 8; ++nt) {
            const int col = nt * 16 + l15;
            bfrag[nt].x = srcXW[r0 * HD + col];
            bfrag[nt].y = srcXW[(r0 + 1) * HD + col];
        }
#pragma unroll
        for (int mt = 0; mt < 2; ++mt)
#pragma unroll
            for (int nt = 0; nt < 8; ++nt)
                acc[mt][nt] = wmma_f32(afrag[mt], bfrag[nt], acc[mt][nt]);
    }

#pragma unroll
    for (int mt = 0; mt < 2; ++mt)
#pragma unroll
        for (int nt = 0; nt < 8; ++nt) {
            const int col = nt * 16 + l15;
            const float bv = bias[col];
#pragma unroll
            for (int v = 0; v < 8; ++v) {
                const int row = M0 + mt * 16 + half * 8 + v;
                float o = acc[mt][nt][v] + bv;
                if (relu) o = fmaxf(o, 0.f);
                if (toGlobal) dstG[(size_t)row * HD + col] = o;
                else          dstLDS[row * ldsStride + col] = o;
            }
        }
}

__global__ void __launch_bounds__(256)
gcn_fused(const float* __restrict__ atoms, const float* __restrict__ adjs,
          const float* __restrict__ W1, const float* __restrict__ b1,
          const float* __restrict__ W2, const float* __restrict__ b2,
          float* __restrict__ out)
{
    extern __shared__ float smem[];
    float* bufA = smem;            // X (stride 76) then H (stride 132): 256*132 floats
    float* bufB = smem + OFF_B;    // XW: 256*128 floats
    float* sdis = smem + OFF_DIS;  // dis[256]

    const int b    = blockIdx.x;
    const int tid  = threadIdx.x;
    const int wave = tid >> 5;
    const int lane = tid & 31;
    const float* adjB = adjs + (size_t)b * NN * NN;

    // ---- Phase 0a: degree / dis (coalesced row sweep; thread c owns column c)
    {
        const int c = tid;
        float deg = 1.f;  // self-loop
        for (int r = 0; r < NN; ++r) {
            const float v = adjB[r * NN + c];
            deg += ((v > 0.5f) && (r != c)) ? 1.f : 0.f;
        }
        sdis[c] = rsqrtf(deg);
    }
    // ---- Phase 0b: stage atoms into LDS, zero-padded K: 75 -> 76
    {
        const float* at = atoms + (size_t)b * NN * AD;
        for (int i = tid; i < NN * KP1; i += 256) {
            const int row = i / KP1, col = i % KP1;
            bufA[row * KP1 + col] = (col < AD) ? at[row * AD + col] : 0.f;
        }
    }
    __syncthreads();

    // ---- Phase 1: XW1 = X @ W1         (LDS -> LDS)
    gemm_xw(bufA, KP1, AD, KP1, W1, bufB, wave, lane);
    __syncthreads();

    // ---- Phase 2: H = relu(A^T @ XW1 + b1)   (-> LDS, stride 132)
    gemm_adj(adjB, sdis, bufB, b1, /*relu=*/true, bufA, HS, nullptr, false, wave, lane);
    __syncthreads();

    // ---- Phase 3: XW2 = H @ W2         (LDS -> LDS)
    gemm_xw(bufA, HS, HD, HD, W2, bufB, wave, lane);
    __syncthreads();

    // ---- Phase 4: out = A^T @ XW2 + b2 (-> global)
    gemm_adj(adjB, sdis, bufB, b2, /*relu=*/false, nullptr, 0,
             out + (size_t)b * NN * HD, true, wave, lane);
}

extern "C" void kernel_launch(void* const* d_in, const int* in_sizes, int n_in,
                              void* d_out, int out_size, void* d_ws, size_t ws_size,
                              hipStream_t stream) {
    (void)in_sizes; (void)n_in; (void)out_size; (void)d_ws; (void)ws_size;
    const float* atoms = (const float*)d_in[0];
    const float* adjs  = (const float*)d_in[1];
    const float* W1    = (const float*)d_in[2];
    const float* b1    = (const float*)d_in[3];
    const float* W2    = (const float*)d_in[4];
    const float* b2    = (const float*)d_in[5];
    float* out = (float*)d_out;

    const size_t shmem = (size_t)SMEM_FLOATS * sizeof(float);  // ~261 KB (<= 320 KB/WGP)
    gcn_fused<<<dim3(NB), dim3(256), shmem, stream>>>(atoms, adjs, W1, b1, W2, b2, out);
}